// Net_12567074308661
// MI455X (gfx1250) — compile-verified
//
#include <hip/hip_runtime.h>
#include <math.h>

typedef __attribute__((ext_vector_type(2))) float v2f;
typedef __attribute__((ext_vector_type(8))) float v8f;

#define N_NODES 100000
#define IN_F    256
#define HID     32
#define NCLS    40

// ---------------------------------------------------------------- utilities

__global__ void zero_kernel(float* __restrict__ p, long n) {
    long i = (long)blockIdx.x * blockDim.x + threadIdx.x;
    long stride = (long)gridDim.x * blockDim.x;
    for (; i < n; i += stride) p[i] = 0.0f;
}

__global__ void deg_kernel(const int* __restrict__ src, const int* __restrict__ dst,
                           float* __restrict__ deg_out, float* __restrict__ deg_in, int E) {
    int i = blockIdx.x * blockDim.x + threadIdx.x;
    int stride = gridDim.x * blockDim.x;
    for (int e = i; e < E; e += stride) {
        atomicAdd(&deg_out[src[e]], 1.0f);
        atomicAdd(&deg_in[dst[e]], 1.0f);
    }
}

__global__ void rsqrt_kernel(float* __restrict__ p, long n) {
    long i = (long)blockIdx.x * blockDim.x + threadIdx.x;
    long stride = (long)gridDim.x * blockDim.x;
    for (; i < n; i += stride) p[i] = rsqrtf(p[i]);  // self loops => deg >= 1
}

// ---------------------------------------------------------------- GEMM 1
// h1 = (X @ W1) * norm_src[row]   (row scale folded in post-accumulation)
// One wave computes a 16-row x 32-col tile with V_WMMA_F32_16X16X4_F32.
__global__ void gemm1_wmma(const float* __restrict__ X, const float* __restrict__ W1,
                           const float* __restrict__ norm_src, float* __restrict__ h1) {
    const int lane  = threadIdx.x & 31;
    const int wave  = (blockIdx.x * blockDim.x + threadIdx.x) >> 5;
    const int row0  = wave * 16;
    if (row0 >= N_NODES) return;             // wave-uniform: EXEC stays all-ones
    const int m     = lane & 15;             // A row / B col within tile
    const int khalf = (lane >> 4) * 2;       // lanes 0-15: K=+0,+1 ; lanes 16-31: K=+2,+3

    v8f acc0 = {};                           // cols 0..15
    v8f acc1 = {};                           // cols 16..31
    const float* arow = X + (size_t)(row0 + m) * IN_F + khalf;

    #pragma unroll 4
    for (int k0 = 0; k0 < IN_F; k0 += 4) {
        v2f a = *(const v2f*)(arow + k0);                    // contiguous b64 load
        const float* bp = W1 + (size_t)(k0 + khalf) * HID + m;
        v2f b0 = { bp[0],  bp[HID]      };                   // {W1[k][n], W1[k+1][n]}
        v2f b1 = { bp[16], bp[HID + 16] };
        acc0 = __builtin_amdgcn_wmma_f32_16x16x4_f32(false, a, false, b0,
                                                     (short)0, acc0, false, false);
        acc1 = __builtin_amdgcn_wmma_f32_16x16x4_f32(false, a, false, b1,
                                                     (short)0, acc1, false, false);
    }

    // C layout: VGPR r -> M = r (lanes 0-15) / M = 8+r (lanes 16-31); N = lane&15
    const int mb = row0 + ((lane >> 4) * 8);
    #pragma unroll
    for (int r = 0; r < 8; ++r) {
        float s = norm_src[mb + r];
        float* o = h1 + (size_t)(mb + r) * HID;
        o[m]      = acc0[r] * s;
        o[16 + m] = acc1[r] * s;
    }
}

// ---------------------------------------------------------------- edge scatter, 32-wide
// One wave per edge: coalesced 128B gather of h1[src], 32 atomic lanes into agg1[dst].
__global__ void edge32_kernel(const int* __restrict__ src, const int* __restrict__ dst,
                              const float* __restrict__ h, float* __restrict__ agg, int E) {
    const int lane = threadIdx.x & 31;
    int wave   = (blockIdx.x * blockDim.x + threadIdx.x) >> 5;
    int nwaves = (gridDim.x * blockDim.x) >> 5;
    for (int e = wave; e < E; e += nwaves) {
        int s = src[e], d = dst[e];
        float v = h[(size_t)s * HID + lane];
        atomicAdd(&agg[(size_t)d * HID + lane], v);
    }
}

// ---------------------------------------------------------------- GEMM 2 (fused)
// x1 = relu(agg1*norm_dst + b1);  h2 = (x1 * norm_src) @ W2   -> [N, 40]
__global__ void gemm2_wmma(const float* __restrict__ agg1, const float* __restrict__ W2,
                           const float* __restrict__ b1, const float* __restrict__ norm_src,
                           const float* __restrict__ norm_dst, float* __restrict__ h2) {
    const int lane  = threadIdx.x & 31;
    const int wave  = (blockIdx.x * blockDim.x + threadIdx.x) >> 5;
    const int row0  = wave * 16;
    if (row0 >= N_NODES) return;
    const int m     = lane & 15;
    const int khalf = (lane >> 4) * 2;
    const int row   = row0 + m;
    const float ns  = norm_src[row];
    const float nd  = norm_dst[row];

    v8f acc0 = {}, acc1 = {}, acc2 = {};     // cols 0..15, 16..31, 32..47(only 32..39 valid)
    #pragma unroll
    for (int k0 = 0; k0 < HID; k0 += 4) {
        int k = k0 + khalf;
        v2f g = *(const v2f*)(agg1 + (size_t)row * HID + k);
        v2f a;
        a.x = fmaxf(fmaf(g.x, nd, b1[k]),     0.0f) * ns;
        a.y = fmaxf(fmaf(g.y, nd, b1[k + 1]), 0.0f) * ns;
        const float* bp = W2 + (size_t)k * NCLS + m;
        v2f b0 = { bp[0],  bp[NCLS]      };
        v2f b1v = { bp[16], bp[NCLS + 16] };
        v2f b2 = { (m < 8) ? bp[32] : 0.0f,
                   (m < 8) ? bp[NCLS + 32] : 0.0f };        // pad cols 40..47 with zeros
        acc0 = __builtin_amdgcn_wmma_f32_16x16x4_f32(false, a, false, b0,
                                                     (short)0, acc0, false, false);
        acc1 = __builtin_amdgcn_wmma_f32_16x16x4_f32(false, a, false, b1v,
                                                     (short)0, acc1, false, false);
        acc2 = __builtin_amdgcn_wmma_f32_16x16x4_f32(false, a, false, b2,
                                                     (short)0, acc2, false, false);
    }

    const int mb = row0 + ((lane >> 4) * 8);
    #pragma unroll
    for (int r = 0; r < 8; ++r) {
        float* o = h2 + (size_t)(mb + r) * NCLS;
        o[m]      = acc0[r];
        o[16 + m] = acc1[r];
        if (m < 8) o[32 + m] = acc2[r];
    }
}

// ---------------------------------------------------------------- edge scatter, 40-wide
__global__ void edge40_kernel(const int* __restrict__ src, const int* __restrict__ dst,
                              const float* __restrict__ h2, float* __restrict__ out, int E) {
    const int lane = threadIdx.x & 31;
    int wave   = (blockIdx.x * blockDim.x + threadIdx.x) >> 5;
    int nwaves = (gridDim.x * blockDim.x) >> 5;
    for (int e = wave; e < E; e += nwaves) {
        int s = src[e], d = dst[e];
        const float* hp = h2 + (size_t)s * NCLS;
        float* op = out + (size_t)d * NCLS;
        atomicAdd(&op[lane], hp[lane]);
        if (lane < 8) atomicAdd(&op[32 + lane], hp[32 + lane]);
    }
}

// ---------------------------------------------------------------- finalize
// out = log_softmax(out*norm_dst + b2) per row; one wave per row (wave32 shuffles).
__global__ void finalize_kernel(float* __restrict__ out, const float* __restrict__ norm_dst,
                                const float* __restrict__ b2) {
    const int lane = threadIdx.x & 31;
    const int row  = (blockIdx.x * blockDim.x + threadIdx.x) >> 5;
    if (row >= N_NODES) return;
    const float nd = norm_dst[row];
    float* op = out + (size_t)row * NCLS;

    float y0 = fmaf(op[lane], nd, b2[lane]);
    float y1 = (lane < 8) ? fmaf(op[32 + lane], nd, b2[32 + lane]) : -__builtin_inff();

    float mx = fmaxf(y0, y1);
    #pragma unroll
    for (int off = 16; off > 0; off >>= 1) mx = fmaxf(mx, __shfl_xor(mx, off, 32));
    float s = expf(y0 - mx) + ((lane < 8) ? expf(y1 - mx) : 0.0f);
    #pragma unroll
    for (int off = 16; off > 0; off >>= 1) s += __shfl_xor(s, off, 32);
    float lse = logf(s) + mx;

    op[lane] = y0 - lse;
    if (lane < 8) op[32 + lane] = y1 - lse;
}

// ---------------------------------------------------------------- launch

extern "C" void kernel_launch(void* const* d_in, const int* in_sizes, int n_in,
                              void* d_out, int out_size, void* d_ws, size_t ws_size,
                              hipStream_t stream) {
    const float* features = (const float*)d_in[0];
    const int*   src      = (const int*)d_in[1];
    const int*   dst      = (const int*)d_in[2];
    const float* W1       = (const float*)d_in[3];
    const float* b1       = (const float*)d_in[4];
    const float* W2       = (const float*)d_in[5];
    const float* b2       = (const float*)d_in[6];
    float*       out      = (float*)d_out;
    const int    E        = in_sizes[1];
    const long   N        = N_NODES;

    // workspace layout (floats): norm_src[N] | norm_dst[N] | h1[32N] | agg1[32N] | h2[40N]
    float* ws       = (float*)d_ws;
    float* norm_src = ws;
    float* norm_dst = ws + N;
    float* h1       = ws + 2 * N;
    float* agg1     = h1 + (size_t)HID * N;
    float* h2       = agg1 + (size_t)HID * N;

    const int T = 256;

    // zero degree accumulators, layer-1 aggregation buffer, and the output accumulator
    zero_kernel<<<2048, T, 0, stream>>>(norm_src, 2 * N);
    zero_kernel<<<2048, T, 0, stream>>>(agg1, (long)HID * N);
    zero_kernel<<<2048, T, 0, stream>>>(out, (long)NCLS * N);

    deg_kernel<<<2048, T, 0, stream>>>(src, dst, norm_src, norm_dst, E);
    rsqrt_kernel<<<(int)((2 * N + T - 1) / T), T, 0, stream>>>(norm_src, 2 * N);

    const int gemmWaves  = (N_NODES + 15) / 16;                 // 6250
    const int gemmBlocks = (gemmWaves * 32 + T - 1) / T;        // 782
    gemm1_wmma<<<gemmBlocks, T, 0, stream>>>(features, W1, norm_src, h1);

    edge32_kernel<<<2048, T, 0, stream>>>(src, dst, h1, agg1, E);

    gemm2_wmma<<<gemmBlocks, T, 0, stream>>>(agg1, W2, b1, norm_src, norm_dst, h2);

    edge40_kernel<<<2048, T, 0, stream>>>(src, dst, h2, out, E);

    const int finBlocks = (int)((N * 32 + T - 1) / T);          // one wave per row
    finalize_kernel<<<finBlocks, T, 0, stream>>>(out, norm_dst, b2);
}